// SAGE_69801808494648
// MI455X (gfx1250) — compile-verified
//
#include <hip/hip_runtime.h>
#include <hip/hip_bf16.h>

typedef float v2f __attribute__((ext_vector_type(2)));
typedef float v8f __attribute__((ext_vector_type(8)));

#define N0_  120000
#define N1_  10000
#define N2_  1024
#define E1_  250000
#define E2_  25600
#define DIN  602
#define DH   256
#define DOUT 41

// ---------------- workspace layouts (float elements) ----------------
// BIG path:   y[N0*DH] | sum1p[N1*DH] | cnt1[N1] | h[N1*DH] | sum2[N2*DH] | cnt2[N2] | logits[N2*DOUT]
#define B_OFF_Y     0ll
#define B_OFF_SUM1P (B_OFF_Y     + (long long)N0_ * DH)
#define B_OFF_CNT1  (B_OFF_SUM1P + (long long)N1_ * DH)
#define B_OFF_H     (B_OFF_CNT1  + N1_)
#define B_OFF_SUM2  (B_OFF_H     + (long long)N1_ * DH)
#define B_OFF_CNT2  (B_OFF_SUM2  + (long long)N2_ * DH)
#define B_OFF_LOG   (B_OFF_CNT2  + N2_)
#define B_TOTAL     (B_OFF_LOG   + (long long)N2_ * DOUT)
// SMALL path: sum1[N1*DIN] | cnt1[N1] | h[N1*DH] | sum2[N2*DH] | cnt2[N2] | logits[N2*DOUT]
#define S_OFF_SUM1  0ll
#define S_OFF_CNT1  (S_OFF_SUM1 + (long long)N1_ * DIN)
#define S_OFF_H     (S_OFF_CNT1 + N1_)
#define S_OFF_SUM2  (S_OFF_H    + (long long)N1_ * DH)
#define S_OFF_CNT2  (S_OFF_SUM2 + (long long)N2_ * DH)
#define S_OFF_LOG   (S_OFF_CNT2 + N2_)

__global__ void zero_f32(float* __restrict__ p, long long n) {
  long long i = (long long)blockIdx.x * blockDim.x + threadIdx.x;
  long long stride = (long long)gridDim.x * blockDim.x;
  for (; i < n; i += stride) p[i] = 0.0f;
}

// One wave (32 lanes) per edge: lanes stride the feature dim; native f32 atomics.
__global__ void scatter_sum(const float* __restrict__ x, const int* __restrict__ src,
                            const int* __restrict__ dst, float* __restrict__ sum,
                            float* __restrict__ cnt, int nedges, int dim) {
  int e = blockIdx.x * (blockDim.x >> 5) + (threadIdx.x >> 5);
  if (e >= nedges) return;
  int lane = threadIdx.x & 31;
  int s = src[e], d = dst[e];
  const float* xs = x + (long long)s * dim;
  float* sd = sum + (long long)d * dim;
  for (int k = lane; k < dim; k += 32)
    unsafeAtomicAdd(&sd[k], xs[k]);
  if (lane == 0) unsafeAtomicAdd(&cnt[d], 1.0f);
}

// ---------------- BIG path kernels ----------------

// y = x @ Wl1 for all N0 rows. One wave computes a full 16x256 strip
// (16 accumulator tiles = 128 VGPRs) so x is streamed from HBM exactly once.
__global__ void __launch_bounds__(64) proj_wmma(
    const float* __restrict__ x, const float* __restrict__ Wl, float* __restrict__ y) {
  int tileM = blockIdx.x * (blockDim.x >> 5) + (threadIdx.x >> 5);
  int lane  = threadIdx.x & 31;
  int half  = lane >> 4;
  int l     = lane & 15;
  int m     = tileM * 16 + l;
  const float* ax = x + (long long)m * DIN;

  v8f c[16];
  #pragma unroll
  for (int t = 0; t < 16; ++t) c[t] = (v8f){};

  for (int k0 = 0; k0 < 600; k0 += 4) {
    int ka = k0 + 2 * half;
    v2f a; a[0] = ax[ka]; a[1] = ax[ka + 1];
    #pragma unroll
    for (int t = 0; t < 16; ++t) {
      int n = t * 16 + l;
      v2f bb;
      bb[0] = Wl[(long long)ka * DH + n];
      bb[1] = Wl[(long long)(ka + 1) * DH + n];
      c[t] = __builtin_amdgcn_wmma_f32_16x16x4_f32(false, a, false, bb, (short)0, c[t], false, false);
    }
  }
  { // K tail: k = 600,601 live only in half==0 lanes
    int ka = 600 + 2 * half;
    v2f a = {0.f, 0.f};
    if (half == 0) { a[0] = ax[ka]; a[1] = ax[ka + 1]; }
    #pragma unroll
    for (int t = 0; t < 16; ++t) {
      int n = t * 16 + l;
      v2f bb = {0.f, 0.f};
      if (half == 0) {
        bb[0] = Wl[(long long)ka * DH + n];
        bb[1] = Wl[(long long)(ka + 1) * DH + n];
      }
      c[t] = __builtin_amdgcn_wmma_f32_16x16x4_f32(false, a, false, bb, (short)0, c[t], false, false);
    }
  }
  #pragma unroll
  for (int t = 0; t < 16; ++t) {
    int n = t * 16 + l;
    #pragma unroll
    for (int r = 0; r < 8; ++r)
      y[(long long)(tileM * 16 + r + 8 * half) * DH + n] = c[t][r];
  }
}

// Hop-1 finalize: h = relu(sum1p/cnt + x_dst @ Wr1 + b).  One 16x16 tile / wave.
__global__ void __launch_bounds__(128) sage1b_wmma(
    const float* __restrict__ sum1p, const float* __restrict__ cnt1,
    const float* __restrict__ x, const float* __restrict__ Wr,
    const float* __restrict__ b, float* __restrict__ h) {
  const int NT = DH / 16;
  int tile  = blockIdx.x * (blockDim.x >> 5) + (threadIdx.x >> 5);
  int tileM = tile / NT, tileN = tile % NT;
  int lane  = threadIdx.x & 31;
  int half  = lane >> 4;
  int l     = lane & 15;
  int m     = tileM * 16 + l;
  int n     = tileN * 16 + l;
  const float* ax = x + (long long)m * DIN;

  v8f c = {};
  for (int k0 = 0; k0 < 600; k0 += 4) {
    int ka = k0 + 2 * half;
    v2f a, bb;
    a[0]  = ax[ka];                     a[1]  = ax[ka + 1];
    bb[0] = Wr[(long long)ka * DH + n]; bb[1] = Wr[(long long)(ka + 1) * DH + n];
    c = __builtin_amdgcn_wmma_f32_16x16x4_f32(false, a, false, bb, (short)0, c, false, false);
  }
  {
    int ka = 600 + 2 * half;
    v2f a = {0.f, 0.f}, bb = {0.f, 0.f};
    if (half == 0) {
      a[0]  = ax[ka];                     a[1]  = ax[ka + 1];
      bb[0] = Wr[(long long)ka * DH + n]; bb[1] = Wr[(long long)(ka + 1) * DH + n];
    }
    c = __builtin_amdgcn_wmma_f32_16x16x4_f32(false, a, false, bb, (short)0, c, false, false);
  }
  float bias = b[n];
  #pragma unroll
  for (int r = 0; r < 8; ++r) {
    int row = tileM * 16 + r + 8 * half;
    float inv = 1.0f / fmaxf(cnt1[row], 1.0f);
    float v = c[r] + sum1p[(long long)row * DH + n] * inv + bias;
    h[(long long)row * DH + n] = fmaxf(v, 0.f);
  }
}

// ---------------- SMALL (fallback) path kernel ----------------

// h = relu((sum1/cnt) @ Wl + x_dst @ Wr + b), dual-GEMM in 602-space.
__global__ void __launch_bounds__(128) sage1_wmma(
    const float* __restrict__ sum1, const float* __restrict__ cnt1,
    const float* __restrict__ x,    const float* __restrict__ Wl,
    const float* __restrict__ Wr,   const float* __restrict__ b,
    float* __restrict__ h) {
  const int NT = DH / 16;
  int tile  = blockIdx.x * (blockDim.x >> 5) + (threadIdx.x >> 5);
  int tileM = tile / NT, tileN = tile % NT;
  int lane  = threadIdx.x & 31;
  int half  = lane >> 4;
  int l     = lane & 15;
  int m     = tileM * 16 + l;
  int n     = tileN * 16 + l;
  const float* as = sum1 + (long long)m * DIN;
  const float* ax = x    + (long long)m * DIN;
  float inv = 1.0f / fmaxf(cnt1[m], 1.0f);

  v8f c = {};
  for (int k0 = 0; k0 < 600; k0 += 4) {
    int ka = k0 + 2 * half;
    v2f aS, aX, bL, bR;
    aS[0] = as[ka] * inv;  aS[1] = as[ka + 1] * inv;
    aX[0] = ax[ka];        aX[1] = ax[ka + 1];
    bL[0] = Wl[(long long)ka * DH + n];  bL[1] = Wl[(long long)(ka + 1) * DH + n];
    bR[0] = Wr[(long long)ka * DH + n];  bR[1] = Wr[(long long)(ka + 1) * DH + n];
    c = __builtin_amdgcn_wmma_f32_16x16x4_f32(false, aX, false, bR, (short)0, c, false, false);
    c = __builtin_amdgcn_wmma_f32_16x16x4_f32(false, aS, false, bL, (short)0, c, false, false);
  }
  {
    int ka = 600 + 2 * half;
    v2f aS = {0.f, 0.f}, aX = {0.f, 0.f}, bL = {0.f, 0.f}, bR = {0.f, 0.f};
    if (half == 0) {
      aS[0] = as[ka] * inv;  aS[1] = as[ka + 1] * inv;
      aX[0] = ax[ka];        aX[1] = ax[ka + 1];
      bL[0] = Wl[(long long)ka * DH + n];  bL[1] = Wl[(long long)(ka + 1) * DH + n];
      bR[0] = Wr[(long long)ka * DH + n];  bR[1] = Wr[(long long)(ka + 1) * DH + n];
    }
    c = __builtin_amdgcn_wmma_f32_16x16x4_f32(false, aX, false, bR, (short)0, c, false, false);
    c = __builtin_amdgcn_wmma_f32_16x16x4_f32(false, aS, false, bL, (short)0, c, false, false);
  }
  float bias = b[n];
  #pragma unroll
  for (int r = 0; r < 8; ++r) {
    float v = c[r] + bias;
    v = v > 0.f ? v : 0.f;
    h[(long long)(tileM * 16 + r + 8 * half) * DH + n] = v;
  }
}

// ---------------- Hop-2 + softmax (shared) ----------------

__global__ void __launch_bounds__(128) sage2_wmma(
    const float* __restrict__ sum2, const float* __restrict__ cnt2,
    const float* __restrict__ h,    const float* __restrict__ Wl,
    const float* __restrict__ Wr,   const float* __restrict__ b,
    float* __restrict__ logits) {
  const int NT = 3;  // ceil(41/16)
  int tile  = blockIdx.x * (blockDim.x >> 5) + (threadIdx.x >> 5);
  int tileM = tile / NT, tileN = tile % NT;
  int lane  = threadIdx.x & 31;
  int half  = lane >> 4;
  int l     = lane & 15;
  int m     = tileM * 16 + l;
  int n     = tileN * 16 + l;
  bool ncol = (n < DOUT);
  const float* as = sum2 + (long long)m * DH;
  const float* ax = h    + (long long)m * DH;
  float inv = 1.0f / fmaxf(cnt2[m], 1.0f);

  v8f c = {};
  for (int k0 = 0; k0 < DH; k0 += 4) {
    int ka = k0 + 2 * half;
    v2f aS, aX, bL = {0.f, 0.f}, bR = {0.f, 0.f};
    aS[0] = as[ka] * inv;  aS[1] = as[ka + 1] * inv;
    aX[0] = ax[ka];        aX[1] = ax[ka + 1];
    if (ncol) {
      bL[0] = Wl[ka * DOUT + n];  bL[1] = Wl[(ka + 1) * DOUT + n];
      bR[0] = Wr[ka * DOUT + n];  bR[1] = Wr[(ka + 1) * DOUT + n];
    }
    c = __builtin_amdgcn_wmma_f32_16x16x4_f32(false, aX, false, bR, (short)0, c, false, false);
    c = __builtin_amdgcn_wmma_f32_16x16x4_f32(false, aS, false, bL, (short)0, c, false, false);
  }
  if (ncol) {
    float bias = b[n];
    #pragma unroll
    for (int r = 0; r < 8; ++r)
      logits[(long long)(tileM * 16 + r + 8 * half) * DOUT + n] = c[r] + bias;
  }
}

__global__ void logsoftmax41(const float* __restrict__ logits, float* __restrict__ out) {
  int row = blockIdx.x * (blockDim.x >> 5) + (threadIdx.x >> 5);
  if (row >= N2_) return;
  int lane = threadIdx.x & 31;
  const float* p = logits + (long long)row * DOUT;
  float v0 = (lane < DOUT)      ? p[lane]      : -3.4e38f;
  float v1 = (lane + 32 < DOUT) ? p[lane + 32] : -3.4e38f;
  float mx = fmaxf(v0, v1);
  for (int o = 16; o > 0; o >>= 1) mx = fmaxf(mx, __shfl_xor(mx, o, 32));
  float s = ((lane < DOUT) ? expf(v0 - mx) : 0.f) +
            ((lane + 32 < DOUT) ? expf(v1 - mx) : 0.f);
  for (int o = 16; o > 0; o >>= 1) s += __shfl_xor(s, o, 32);
  float lse = mx + logf(s);
  if (lane < DOUT)      out[(long long)row * DOUT + lane]      = v0 - lse;
  if (lane + 32 < DOUT) out[(long long)row * DOUT + lane + 32] = v1 - lse;
}

extern "C" void kernel_launch(void* const* d_in, const int* in_sizes, int n_in,
                              void* d_out, int out_size, void* d_ws, size_t ws_size,
                              hipStream_t stream) {
  const float* x   = (const float*)d_in[0];
  const int*   e1s = (const int*)  d_in[1];
  const int*   e1d = (const int*)  d_in[2];
  const int*   e2s = (const int*)  d_in[3];
  const int*   e2d = (const int*)  d_in[4];
  const float* Wl1 = (const float*)d_in[5];
  const float* Wr1 = (const float*)d_in[6];
  const float* bl1 = (const float*)d_in[7];
  const float* Wl2 = (const float*)d_in[8];
  const float* Wr2 = (const float*)d_in[9];
  const float* bl2 = (const float*)d_in[10];
  float* ws  = (float*)d_ws;
  float* out = (float*)d_out;

  if (ws_size >= (size_t)B_TOTAL * sizeof(float)) {
    // ---- BIG path: project-then-scatter (256-wide scatter instead of 602) ----
    float* y      = ws + B_OFF_Y;
    float* sum1p  = ws + B_OFF_SUM1P;
    float* cnt1   = ws + B_OFF_CNT1;
    float* h      = ws + B_OFF_H;
    float* sum2   = ws + B_OFF_SUM2;
    float* cnt2   = ws + B_OFF_CNT2;
    float* logits = ws + B_OFF_LOG;

    zero_f32<<<2048, 256, 0, stream>>>(sum1p, (long long)N1_ * DH + N1_);
    zero_f32<<<512,  256, 0, stream>>>(sum2,  (long long)N2_ * DH + N2_);

    proj_wmma<<<(N0_ / 16) / 2, 64, 0, stream>>>(x, Wl1, y);                     // y = x @ Wl1
    scatter_sum<<<(E1_ + 7) / 8, 256, 0, stream>>>(y, e1s, e1d, sum1p, cnt1, E1_, DH);
    sage1b_wmma<<<(N1_ / 16) * (DH / 16) / 4, 128, 0, stream>>>(sum1p, cnt1, x, Wr1, bl1, h);

    scatter_sum<<<(E2_ + 7) / 8, 256, 0, stream>>>(h, e2s, e2d, sum2, cnt2, E2_, DH);
    sage2_wmma<<<(N2_ / 16) * 3 / 4, 128, 0, stream>>>(sum2, cnt2, h, Wl2, Wr2, bl2, logits);
    logsoftmax41<<<(N2_ + 7) / 8, 256, 0, stream>>>(logits, out);
  } else {
    // ---- SMALL path: direct 602-wide scatter + dual GEMM ----
    float* sum1   = ws + S_OFF_SUM1;
    float* cnt1   = ws + S_OFF_CNT1;
    float* h      = ws + S_OFF_H;
    float* sum2   = ws + S_OFF_SUM2;
    float* cnt2   = ws + S_OFF_CNT2;
    float* logits = ws + S_OFF_LOG;

    zero_f32<<<4096, 256, 0, stream>>>(sum1, (long long)N1_ * DIN + N1_);
    zero_f32<<<512,  256, 0, stream>>>(sum2, (long long)N2_ * DH  + N2_);

    scatter_sum<<<(E1_ + 7) / 8, 256, 0, stream>>>(x, e1s, e1d, sum1, cnt1, E1_, DIN);
    sage1_wmma<<<(N1_ / 16) * (DH / 16) / 4, 128, 0, stream>>>(sum1, cnt1, x, Wl1, Wr1, bl1, h);

    scatter_sum<<<(E2_ + 7) / 8, 256, 0, stream>>>(h, e2s, e2d, sum2, cnt2, E2_, DH);
    sage2_wmma<<<(N2_ / 16) * 3 / 4, 128, 0, stream>>>(sum2, cnt2, h, Wl2, Wr2, bl2, logits);
    logsoftmax41<<<(N2_ + 7) / 8, 256, 0, stream>>>(logits, out);
  }
}